// CustomLoss_61460982006014
// MI455X (gfx1250) — compile-verified
//
#include <hip/hip_runtime.h>
#include <math.h>

typedef float v2f __attribute__((ext_vector_type(2)));
typedef float v8f __attribute__((ext_vector_type(8)));

#define WAVE 32

// Force a wave-uniform value into an SGPR so guards become scalar branches
// (keeps EXEC all-1s around WMMA, removes v_cmpx/exec save-restore overhead).
__device__ __forceinline__ int wave_uniform(int x) {
    return __builtin_amdgcn_readfirstlane(x);
}

// ---------------------------------------------------------------------------
// Sum-of-squares reduction using V_WMMA_F32_16X16X4_F32 with an all-ones B:
//   D[i][j] += sum_k A[i][k] * 1.0   ->  row sums of A accumulate in C.
// Feed A = x^2, keep C live across the whole grid-stride loop, extract at end.
// MODE 0: x = a[i] - b[i]   (streamed non-temporally; do not pollute L2)
// MODE 1: x = max(a[i], 0)  (regular loads; keep E/W resident in L2)
// ---------------------------------------------------------------------------
template <int MODE>
__global__ void sumsq_kernel(const float* __restrict__ a,
                             const float* __restrict__ b,
                             long n, float* __restrict__ partials)
{
    const unsigned lane          = threadIdx.x & (WAVE - 1);
    const unsigned waveInBlock   = threadIdx.x / WAVE;      // wave-uniform
    const unsigned wavesPerBlock = blockDim.x / WAVE;
    const int waveGlobal = wave_uniform((int)(blockIdx.x * wavesPerBlock + waveInBlock));
    const int totalWaves = (int)(gridDim.x * wavesPerBlock);

    const int nChunks = (int)(n / 64);   // 64 f32 per 16x4 A tile per wave
    // element offset inside a 64-elem chunk for this lane's float2
    // (32-bit A-matrix 16x4 layout: lanes 0-15 hold K=0,1; lanes 16-31 hold K=2,3)
    const unsigned laneOff = (lane & 15u) * 4u + (lane >> 4) * 2u;

    v8f c = {};                 // 16x16 f32 accumulator tile
    v2f onesB = {1.0f, 1.0f};   // B = all ones (4x16, 2 VGPRs)

    const int U = 8;            // chunks per iteration: 2 KB/array/wave in flight
    for (int chunk = waveGlobal * U; chunk < nChunks; chunk += totalWaves * U) {
        if (chunk + U <= nChunks) {                  // scalar test -> s_cbranch
            v2f av[U], bv[U];
#pragma unroll
            for (int u = 0; u < U; ++u) {            // issue all loads first
                long base = (long)(chunk + u) * 64 + laneOff;
                if (MODE == 0) {
                    av[u] = __builtin_nontemporal_load((const v2f*)(a + base));
                    bv[u] = __builtin_nontemporal_load((const v2f*)(b + base));
                } else {
                    av[u] = *(const v2f*)(a + base);
                }
            }
#pragma unroll
            for (int u = 0; u < U; ++u) {            // then compute + accumulate
                v2f x;
                if (MODE == 0) {
                    x.x = av[u].x - bv[u].x;
                    x.y = av[u].y - bv[u].y;
                } else {
                    x.x = fmaxf(av[u].x, 0.0f);
                    x.y = fmaxf(av[u].y, 0.0f);
                }
                v2f sq;
                sq.x = x.x * x.x;
                sq.y = x.y * x.y;
                c = __builtin_amdgcn_wmma_f32_16x16x4_f32(
                        false, sq, false, onesB, (short)0, c, false, false);
            }
        } else {                                     // remainder group (scalar guards)
#pragma unroll
            for (int u = 0; u < U; ++u) {
                int ch = chunk + u;
                if (ch < nChunks) {
                    long base = (long)ch * 64 + laneOff;
                    v2f x;
                    if (MODE == 0) {
                        v2f av = __builtin_nontemporal_load((const v2f*)(a + base));
                        v2f bv = __builtin_nontemporal_load((const v2f*)(b + base));
                        x.x = av.x - bv.x;
                        x.y = av.y - bv.y;
                    } else {
                        v2f av = *(const v2f*)(a + base);
                        x.x = fmaxf(av.x, 0.0f);
                        x.y = fmaxf(av.y, 0.0f);
                    }
                    v2f sq;
                    sq.x = x.x * x.x;
                    sq.y = x.y * x.y;
                    c = __builtin_amdgcn_wmma_f32_16x16x4_f32(
                            false, sq, false, onesB, (short)0, c, false, false);
                }
            }
        }
    }

    // Columns of D are identical. Lanes 0-15 hold rows 0-7 in c[0..7],
    // lanes 16-31 hold rows 8-15 -> one exact xor-16 shuffle gives the total.
    float local = c[0] + c[1] + c[2] + c[3] + c[4] + c[5] + c[6] + c[7];
    local += __shfl_xor(local, 16, WAVE);

    __shared__ float smem[32];
    if (lane == 0) smem[waveInBlock] = local;
    __syncthreads();
    if (threadIdx.x == 0) {
        float p = 0.0f;
        for (unsigned w = 0; w < wavesPerBlock; ++w) p += smem[w];
        if (blockIdx.x == 0) {                       // scalar tail (n % 64)
            for (long i = (long)nChunks * 64; i < n; ++i) {
                float x = (MODE == 0) ? (a[i] - b[i]) : fmaxf(a[i], 0.0f);
                p += x * x;
            }
        }
        partials[blockIdx.x] = p;
    }
}

// ---------------------------------------------------------------------------
// Weighted pair-distance penalty: one wave per (b, j) pair, K = 64 (2 per lane).
// P element (idx, k) is at P[idx*strideIdx + k*strideK]:
//   E rows:    strideIdx = K,       strideK = 1        (contiguous, L2-hot)
//   W columns: strideIdx = 1,       strideK = N_WORD   (L2-resident gathers)
// ---------------------------------------------------------------------------
__global__ void pair_penalty_kernel(const float* __restrict__ P,
                                    long strideK, long strideIdx,
                                    const int* __restrict__ centerIdx,
                                    const int* __restrict__ nbIdx,
                                    const float* __restrict__ Svals,
                                    int nB, int nNb,
                                    float* __restrict__ partials)
{
    const unsigned lane          = threadIdx.x & (WAVE - 1);
    const unsigned waveInBlock   = threadIdx.x / WAVE;
    const unsigned wavesPerBlock = blockDim.x / WAVE;
    const int waveGlobal = wave_uniform((int)(blockIdx.x * wavesPerBlock + waveInBlock));
    const int totalWaves = (int)(gridDim.x * wavesPerBlock);
    const int nPairs = nB * nNb;

    float acc = 0.0f;
    for (int pair = waveGlobal; pair < nPairs; pair += totalWaves) {
        int bi = pair / nNb;                           // scalar
        int ci = wave_uniform(centerIdx[bi]);          // scalar gather indices
        int nj = wave_uniform(nbIdx[pair]);
        const float* pi = P + (long)ci * strideIdx;
        const float* pj = P + (long)nj * strideIdx;
        long k0 = 2 * (long)lane;
        float d0 = pi[k0 * strideK]       - pj[k0 * strideK];
        float d1 = pi[(k0 + 1) * strideK] - pj[(k0 + 1) * strideK];
        float s = d0 * d0 + d1 * d1;
#pragma unroll
        for (int m = 16; m >= 1; m >>= 1) s += __shfl_xor(s, m, WAVE);
        acc += sqrtf(s) * Svals[pair];     // identical in every lane
    }

    __shared__ float smem[32];
    if (lane == 0) smem[waveInBlock] = acc;
    __syncthreads();
    if (threadIdx.x == 0) {
        float p = 0.0f;
        for (unsigned w = 0; w < wavesPerBlock; ++w) p += smem[w];
        partials[blockIdx.x] = p;
    }
}

// ---------------------------------------------------------------------------
// Deterministic final combine: fixed-shape tree reduction of 5 partial
// segments, then loss = sqrt(diff) + r0*ent + r1*word + r2*(sqrt(E)+sqrt(W)).
// ---------------------------------------------------------------------------
__global__ void finalize_kernel(const float* __restrict__ ws,
                                int offDiff, int nDiff,
                                int offE,    int nE,
                                int offW,    int nW,
                                int offEnt,  int nEnt,
                                int offWord, int nWord,
                                const float* __restrict__ reg,
                                float* __restrict__ out)
{
    __shared__ float smem[256];
    __shared__ float segRes[5];
    const int offs[5] = {offDiff, offE, offW, offEnt, offWord};
    const int cnts[5] = {nDiff, nE, nW, nEnt, nWord};

    for (int s = 0; s < 5; ++s) {
        float acc = 0.0f;
        for (int i = threadIdx.x; i < cnts[s]; i += blockDim.x)
            acc += ws[offs[s] + i];
        smem[threadIdx.x] = acc;
        __syncthreads();
        for (int step = blockDim.x / 2; step > 0; step >>= 1) {
            if ((int)threadIdx.x < step) smem[threadIdx.x] += smem[threadIdx.x + step];
            __syncthreads();
        }
        if (threadIdx.x == 0) segRes[s] = smem[0];
        __syncthreads();
    }

    if (threadIdx.x == 0) {
        float loss = sqrtf(segRes[0])
                   + reg[0] * segRes[3]
                   + reg[1] * segRes[4]
                   + reg[2] * (sqrtf(segRes[1]) + sqrtf(segRes[2]));
        out[0] = loss;
    }
}

// ---------------------------------------------------------------------------
extern "C" void kernel_launch(void* const* d_in, const int* in_sizes, int n_in,
                              void* d_out, int out_size, void* d_ws, size_t ws_size,
                              hipStream_t stream)
{
    const float* target        = (const float*)d_in[0];
    const float* prediction    = (const float*)d_in[1];
    const float* reg           = (const float*)d_in[2];
    const float* E             = (const float*)d_in[3];
    const float* W             = (const float*)d_in[4];
    const float* Se_values     = (const float*)d_in[5];
    const float* Sw_values     = (const float*)d_in[6];
    const int*   batch_entities= (const int*)d_in[7];
    const int*   batch_words   = (const int*)d_in[8];
    const int*   Se_indices    = (const int*)d_in[9];
    const int*   Sw_indices    = (const int*)d_in[10];

    const long nBig   = (long)in_sizes[0];   // B * N_WORD = 122,880,000
    const long nEelem = (long)in_sizes[3];   // N_ENT * K  =  12,800,000
    const long nWelem = (long)in_sizes[4];   // K * N_WORD =   1,920,000
    const int  Bn     = in_sizes[7];         // 4096
    const int  nNb    = in_sizes[9] / Bn;    // 32
    const long K      = 64;
    const long N_WORD = nBig / Bn;           // 30000

    // workspace partial-sum layout (floats)
    const int G_DIFF = 1024, G_E = 512, G_W = 256;
    const int G_PAIR = (Bn * nNb) / 8;       // 8 waves/block, one wave per pair
    const int offDiff = 0;
    const int offE    = offDiff + G_DIFF;
    const int offW    = offE    + G_E;
    const int offEnt  = offW    + G_W;
    const int offWord = offEnt  + G_PAIR;
    float* ws = (float*)d_ws;

    const dim3 blk(256);

    // 1) stream E and W first so they are L2-resident for the gather kernels
    sumsq_kernel<1><<<dim3(G_E), blk, 0, stream>>>(E, nullptr, nEelem, ws + offE);
    sumsq_kernel<1><<<dim3(G_W), blk, 0, stream>>>(W, nullptr, nWelem, ws + offW);

    // 2) pair penalties (gathers hit L2: E = 51 MB, W = 7.7 MB << 192 MB)
    pair_penalty_kernel<<<dim3(G_PAIR), blk, 0, stream>>>(
        E, /*strideK=*/1, /*strideIdx=*/K,
        batch_entities, Se_indices, Se_values, Bn, nNb, ws + offEnt);
    pair_penalty_kernel<<<dim3(G_PAIR), blk, 0, stream>>>(
        W, /*strideK=*/N_WORD, /*strideIdx=*/1,
        batch_words, Sw_indices, Sw_values, Bn, nNb, ws + offWord);

    // 3) the ~983 MB streaming term (non-temporal loads)
    sumsq_kernel<0><<<dim3(G_DIFF), blk, 0, stream>>>(target, prediction, nBig,
                                                      ws + offDiff);

    // 4) deterministic combine
    finalize_kernel<<<dim3(1), blk, 0, stream>>>(
        ws, offDiff, G_DIFF, offE, G_E, offW, G_W,
        offEnt, G_PAIR, offWord, G_PAIR, reg, (float*)d_out);
}